// Simple_Cross_Attention_84550726189534
// MI455X (gfx1250) — compile-verified
//
#include <hip/hip_runtime.h>
#include <hip/hip_bf16.h>
#include <math.h>

typedef __attribute__((ext_vector_type(2))) float v2f;
typedef __attribute__((ext_vector_type(4))) float v4f;
typedef __attribute__((ext_vector_type(8))) float v8f;

#define BB 16
#define NN 8192
#define DD 768
#define HH 3072
#define CHUNKS 32              // blocks per batch row in pass1
#define ROWS (NN / CHUNKS)     // 256 rows per block
#define WAVES 8                // 256 threads / wave32
#define V4PL (DD / 128)        // 6 float4 per lane along D

__device__ __forceinline__ float wave_sum32(float v) {
    #pragma unroll
    for (int off = 16; off > 0; off >>= 1) v += __shfl_xor(v, off, 32);
    return v;
}

// ---------------- Pass 1: logits + online-softmax pooled partials ------------
// Streams x once (402 MB) with b128 non-temporal loads; lane l owns
// d = 4*l + 128*j, j = 0..5.
__global__ void pass1_attn_pool(const float* __restrict__ x,
                                const float* __restrict__ query,
                                float* __restrict__ attn_out,
                                float* __restrict__ partM,
                                float* __restrict__ partS,
                                float* __restrict__ partP) {
    const int t    = threadIdx.x;
    const int lane = t & 31;
    const int w    = t >> 5;
    const int blk  = blockIdx.x;           // = b*CHUNKS + c
    const int b    = blk / CHUNKS;
    const int c    = blk % CHUNKS;
    const int n0   = c * ROWS;

    // cache query slice for this lane (reused, regular loads)
    const v4f* q4 = (const v4f*)query;
    v4f qv[V4PL];
    #pragma unroll
    for (int j = 0; j < V4PL; ++j) qv[j] = q4[lane + 32 * j];

    float mw = -INFINITY, sumw = 0.0f;
    v4f acc[V4PL];
    #pragma unroll
    for (int j = 0; j < V4PL; ++j) acc[j] = (v4f){0.f, 0.f, 0.f, 0.f};

    const float DS = 0.03608439182435161f;   // 768^-0.5

    // each wave handles ROWS/WAVES = 32 rows
    for (int i = 0; i < ROWS / WAVES; ++i) {
        const int n = n0 + w + WAVES * i;
        const v4f* xr4 = (const v4f*)(x + ((size_t)b * NN + n) * DD);
        v4f xv[V4PL];
        float dot = 0.0f;
        #pragma unroll
        for (int j = 0; j < V4PL; ++j) {
            xv[j] = __builtin_nontemporal_load(xr4 + lane + 32 * j);  // b128, NT
            dot = fmaf(xv[j].x, qv[j].x, dot);
            dot = fmaf(xv[j].y, qv[j].y, dot);
            dot = fmaf(xv[j].z, qv[j].z, dot);
            dot = fmaf(xv[j].w, qv[j].w, dot);
        }
        dot = wave_sum32(dot);
        if (lane == 0) attn_out[(size_t)b * NN + n] = dot;   // raw logit output

        const float sc   = dot * DS;
        const float mnew = fmaxf(mw, sc);
        const float f    = __expf(mw - mnew);   // 0 on first iter (mw = -inf)
        const float p    = __expf(sc - mnew);
        sumw = sumw * f + p;
        #pragma unroll
        for (int j = 0; j < V4PL; ++j) {
            acc[j].x = fmaf(acc[j].x, f, p * xv[j].x);
            acc[j].y = fmaf(acc[j].y, f, p * xv[j].y);
            acc[j].z = fmaf(acc[j].z, f, p * xv[j].z);
            acc[j].w = fmaf(acc[j].w, f, p * xv[j].w);
        }
        mw = mnew;
    }

    // combine the 8 waves of this block
    __shared__ float wm[WAVES], wsum[WAVES];
    __shared__ float pool[DD];
    for (int d = t; d < DD; d += 256) pool[d] = 0.0f;
    if (lane == 0) { wm[w] = mw; wsum[w] = sumw; }
    __syncthreads();

    float M = wm[0];
    #pragma unroll
    for (int j = 1; j < WAVES; ++j) M = fmaxf(M, wm[j]);
    const float fw = __expf(mw - M);
    #pragma unroll
    for (int j = 0; j < V4PL; ++j) {
        const int d = 4 * lane + 128 * j;
        atomicAdd(&pool[d + 0], acc[j].x * fw);
        atomicAdd(&pool[d + 1], acc[j].y * fw);
        atomicAdd(&pool[d + 2], acc[j].z * fw);
        atomicAdd(&pool[d + 3], acc[j].w * fw);
    }
    __syncthreads();

    if (t == 0) {
        float S = 0.0f;
        #pragma unroll
        for (int j = 0; j < WAVES; ++j) S += wsum[j] * __expf(wm[j] - M);
        partM[blk] = M;
        partS[blk] = S;
    }
    for (int d = t; d < DD; d += 256) partP[(size_t)blk * DD + d] = pool[d];
}

// ---------------- Pass 2: merge partials, normalize, LayerNorm --------------
__global__ void pass2_combine_ln(const float* __restrict__ partM,
                                 const float* __restrict__ partS,
                                 const float* __restrict__ partP,
                                 const float* __restrict__ gamma,
                                 const float* __restrict__ beta,
                                 float* __restrict__ pooled,
                                 float* __restrict__ hln) {
    const int b = blockIdx.x, t = threadIdx.x;
    __shared__ float sm[CHUNKS], sscale[CHUNKS], ssum[CHUNKS];
    __shared__ float red[256];

    if (t < CHUNKS) sm[t] = partM[b * CHUNKS + t];
    __syncthreads();
    float M = sm[0];
    #pragma unroll
    for (int j = 1; j < CHUNKS; ++j) M = fmaxf(M, sm[j]);
    if (t < CHUNKS) {
        sscale[t] = __expf(sm[t] - M);
        ssum[t]   = partS[b * CHUNKS + t] * sscale[t];
    }
    __syncthreads();
    float total = 0.0f;
    #pragma unroll
    for (int j = 0; j < CHUNKS; ++j) total += ssum[j];
    const float inv = 1.0f / total;

    float pv[DD / 256];
    float lsum = 0.0f;
    #pragma unroll
    for (int i = 0; i < DD / 256; ++i) {
        const int d = t + i * 256;
        float a = 0.0f;
        for (int c = 0; c < CHUNKS; ++c)
            a = fmaf(partP[(size_t)(b * CHUNKS + c) * DD + d], sscale[c], a);
        pv[i] = a * inv;
        pooled[b * DD + d] = pv[i];
        lsum += pv[i];
    }
    // mean
    red[t] = lsum; __syncthreads();
    for (int s = 128; s > 0; s >>= 1) { if (t < s) red[t] += red[t + s]; __syncthreads(); }
    const float mu = red[0] * (1.0f / DD);
    __syncthreads();
    // variance
    float lvar = 0.0f;
    #pragma unroll
    for (int i = 0; i < DD / 256; ++i) { const float dl = pv[i] - mu; lvar = fmaf(dl, dl, lvar); }
    red[t] = lvar; __syncthreads();
    for (int s = 128; s > 0; s >>= 1) { if (t < s) red[t] += red[t + s]; __syncthreads(); }
    const float rstd = rsqrtf(red[0] * (1.0f / DD) + 1e-6f);

    #pragma unroll
    for (int i = 0; i < DD / 256; ++i) {
        const int d = t + i * 256;
        hln[b * DD + d] = (pv[i] - mu) * rstd * gamma[d] + beta[d];
    }
}

// ---------------- Pass 3: GEMM1 [16,768]@[768,3072] + bias + GeLU -> y1^T ---
__global__ void gemm1_wmma(const float* __restrict__ hln,
                           const float* __restrict__ w1,
                           const float* __restrict__ b1,
                           float* __restrict__ y1t) {
    __shared__ float A[16 * 769];            // padded, stride 769: conflict-free
    const int t = threadIdx.x;
    for (int i = t; i < 16 * DD; i += 128) {
        const int m = i / DD, k = i % DD;
        A[m * 769 + k] = hln[m * DD + k];
    }
    __syncthreads();

    const int wave  = t >> 5, lane = t & 31;
    const int nbase = blockIdx.x * 64 + wave * 16;
    const int half  = lane >> 4;             // K sub-pair select
    const int mn    = lane & 15;             // M for A-frag, N for B-frag

    v8f cfr = {0.f, 0.f, 0.f, 0.f, 0.f, 0.f, 0.f, 0.f};
    for (int k0 = 0; k0 < DD; k0 += 4) {
        const int kb = k0 + half * 2;
        v2f a, bm;
        a[0]  = A[mn * 769 + kb];
        a[1]  = A[mn * 769 + kb + 1];
        bm[0] = w1[(size_t)kb * HH + nbase + mn];
        bm[1] = w1[(size_t)(kb + 1) * HH + nbase + mn];
        cfr = __builtin_amdgcn_wmma_f32_16x16x4_f32(false, a, false, bm,
                                                    (short)0, cfr, false, false);
    }

    const int n = nbase + mn;
    const float bias = b1[n];
    #pragma unroll
    for (int r = 0; r < 8; ++r) {
        const int m = r + half * 8;
        const float v = cfr[r] + bias;
        const float g = 0.5f * v * (1.0f + erff(v * 0.70710678118654752f));  // exact gelu
        y1t[n * 16 + m] = g;                 // store transposed [H,16]
    }
}

// ---------------- Pass 4: GEMM2 [16,3072]@[3072,768] + b2 + pooled -> out ---
__global__ void gemm2_wmma(const float* __restrict__ y1t,
                           const float* __restrict__ w2,
                           const float* __restrict__ b2,
                           const float* __restrict__ pooled,
                           float* __restrict__ out) {
    const int t = threadIdx.x;
    const int wave  = t >> 5, lane = t & 31;
    const int nbase = blockIdx.x * 64 + wave * 16;
    const int half  = lane >> 4;
    const int mn    = lane & 15;

    v8f cfr = {0.f, 0.f, 0.f, 0.f, 0.f, 0.f, 0.f, 0.f};
    for (int k0 = 0; k0 < HH; k0 += 4) {
        const int kb = k0 + half * 2;
        v2f a, bm;
        a[0]  = y1t[kb * 16 + mn];           // contiguous per half-wave
        a[1]  = y1t[(kb + 1) * 16 + mn];
        bm[0] = w2[(size_t)kb * DD + nbase + mn];
        bm[1] = w2[(size_t)(kb + 1) * DD + nbase + mn];
        cfr = __builtin_amdgcn_wmma_f32_16x16x4_f32(false, a, false, bm,
                                                    (short)0, cfr, false, false);
    }

    const int n = nbase + mn;
    const float bias = b2[n];
    #pragma unroll
    for (int r = 0; r < 8; ++r) {
        const int m = r + half * 8;
        out[m * DD + n] = cfr[r] + bias + pooled[m * DD + n];
    }
}

extern "C" void kernel_launch(void* const* d_in, const int* in_sizes, int n_in,
                              void* d_out, int out_size, void* d_ws, size_t ws_size,
                              hipStream_t stream) {
    const float* x     = (const float*)d_in[0];
    const float* query = (const float*)d_in[1];
    const float* gamma = (const float*)d_in[2];
    const float* beta  = (const float*)d_in[3];
    const float* w1    = (const float*)d_in[4];
    const float* b1    = (const float*)d_in[5];
    const float* w2    = (const float*)d_in[6];
    const float* b2    = (const float*)d_in[7];

    float* out  = (float*)d_out;             // [16,768]
    float* attn = out + BB * DD;             // [16,8192] raw logits

    float* wsf    = (float*)d_ws;
    float* partM  = wsf;                               // B*CHUNKS
    float* partS  = partM + BB * CHUNKS;               // B*CHUNKS
    float* partP  = partS + BB * CHUNKS;               // B*CHUNKS*D
    float* pooled = partP + (size_t)BB * CHUNKS * DD;  // B*D
    float* hln    = pooled + BB * DD;                  // B*D
    float* y1t    = hln + BB * DD;                     // H*16 (transposed)

    pass1_attn_pool<<<BB * CHUNKS, 256, 0, stream>>>(x, query, attn, partM, partS, partP);
    pass2_combine_ln<<<BB, 256, 0, stream>>>(partM, partS, partP, gamma, beta, pooled, hln);
    gemm1_wmma<<<HH / 64, 128, 0, stream>>>(hln, w1, b1, y1t);
    gemm2_wmma<<<DD / 64, 128, 0, stream>>>(y1t, w2, b2, pooled, out);
}